// OctreeMaxPool_51677046505682
// MI455X (gfx1250) — compile-verified
//
#include <hip/hip_runtime.h>
#include <stdint.h>

// ---------------------------------------------------------------------------
// OctreeMaxPool for MI455X (gfx1250)
//   pooled[c,g] = max(data[c, 8g .. 8g+7])            (512MB streaming read)
//   out[c,m]    = idx[m] >= 0 ? pooled[c, idx[m]] : 0 (L2-served gather)
// Kernel 1 stages data via Tensor Data Mover (double-buffered LDS tiles).
// Kernel 2 gathers from L2-resident pooled with non-temporal output stores.
// ---------------------------------------------------------------------------

#define TM 4          // channel rows per TDM tile
#define TK 2048       // columns (sibling elements) per TDM tile
#define TILE_ELEMS (TM * TK)
#define TILE_BYTES (TILE_ELEMS * 4)   // 32 KB per buffer, 64 KB double-buffered

typedef unsigned int u32x4 __attribute__((ext_vector_type(4)));
typedef int          i32x4 __attribute__((ext_vector_type(4)));
typedef int          i32x8 __attribute__((ext_vector_type(8)));
typedef float        f32x4 __attribute__((ext_vector_type(4)));

// The TDM engine writes LDS behind the compiler's back. Escape the buffer
// address into an opaque asm with a memory clobber so LLVM must assume the
// LDS contents changed (otherwise it folds reads of never-stored LDS to
// undef and deletes every ds_load — observed as ds=0 in round 2).
__device__ __forceinline__ void lds_dma_publish(const void* p) {
    asm volatile("" : : "r"(p) : "memory");
}

// Issue one TDM load: TM rows x TK floats, row stride = tensor row stride N,
// packed contiguously into LDS at byte offset lds_off.
__device__ __forceinline__ void tdm_load_tile(const float* gptr, uint32_t lds_off,
                                              uint32_t row_stride_elems) {
    uint64_t ga = (uint64_t)(uintptr_t)gptr;

    // D# group 0 (ISA 8.3): count=1 | lds_addr | global_addr | type=2
    u32x4 g0 = {
        1u,                                                   // count=1 (valid, user)
        lds_off,                                              // lds_addr (bytes)
        (uint32_t)ga,                                         // global_addr[31:0]
        ((uint32_t)(ga >> 32) & 0x01FFFFFFu) | (2u << 30)     // global_addr[56:32] | type=2
    };

    // D# group 1 (ISA 8.4): data_size=4B; tensor dims == tile dims (no OOB);
    // dim0 stride = row stride (elements).
    i32x8 g1 = {
        (int)(2u << 16),                       // workgroup_mask=0, data_size=2 (4B)
        (int)((uint32_t)TK << 16),             // [63:48] tensor_dim0[15:0] = TK
        (int)((uint32_t)TM << 16),             // [95:80] tensor_dim1[15:0] = TM
        (int)((uint32_t)TK << 16),             // [127:112] tile_dim0 = TK
        (int)TM,                               // [143:128] tile_dim1 = TM, tile_dim2=0
        (int)row_stride_elems,                 // tensor_dim0_stride[31:0]
        0,                                     // stride[47:32]=0, dim1_stride[15:0]=0
        0
    };
    i32x4 g2 = {0, 0, 0, 0};                   // 2D tensor: groups 2/3 unused
    i32x4 g3 = {0, 0, 0, 0};

#if defined(__clang_major__) && (__clang_major__ >= 23)
    i32x8 gpad = {0, 0, 0, 0, 0, 0, 0, 0};
    __builtin_amdgcn_tensor_load_to_lds(g0, g1, g2, g3, gpad, 0);
#else
    __builtin_amdgcn_tensor_load_to_lds(g0, g1, g2, g3, 0);
#endif
}

__device__ __forceinline__ float max8_v(f32x4 a, f32x4 b) {
    return fmaxf(fmaxf(fmaxf(a.x, a.y), fmaxf(a.z, a.w)),
                 fmaxf(fmaxf(b.x, b.y), fmaxf(b.z, b.w)));
}

// ---------------------------------------------------------------------------
// Kernel 1: TDM-staged 8:1 max pool.
// Grid: N/TK workgroups of 256 threads; each iterates over C/TM channel tiles.
// ---------------------------------------------------------------------------
__global__ __launch_bounds__(256) void
OctreePool_tdm_kernel(const float* __restrict__ data, float* __restrict__ pooled,
                      uint32_t N, uint32_t G, uint32_t niter) {
    __shared__ __align__(16) float buf[2][TILE_ELEMS];   // only LDS object -> offset 0

    const uint32_t n0  = blockIdx.x * TK;
    const uint32_t tid = threadIdx.x;
    const bool wave0   = (tid < 32);   // wave-uniform: TENSORcnt is per-wave

    // Prologue: prime both buffers (niter >= 2 guaranteed by host).
    if (wave0) {
        tdm_load_tile(data + (uint64_t)0 * N + n0, 0u, N);
        tdm_load_tile(data + (uint64_t)TM * N + n0, (uint32_t)TILE_BYTES, N);
    }

    for (uint32_t it = 0; it < niter; ++it) {
        if (wave0) {
            if (it + 1u < niter) __builtin_amdgcn_s_wait_tensorcnt(1);  // tile `it` done
            else                 __builtin_amdgcn_s_wait_tensorcnt(0);  // last tile done
        }
        __syncthreads();        // publish LDS tile to all waves
        lds_dma_publish(buf);   // compiler-visible "LDS was written by DMA"

        const uint32_t c0 = it * TM;
        const uint32_t b  = it & 1u;
#pragma unroll
        for (uint32_t k = 0; k < TM; ++k) {
            // row k of tile: 256 groups of 8; one group per thread
            const f32x4* p = (const f32x4*)&buf[b][k * TK + tid * 8];
            float v = max8_v(p[0], p[1]);
            pooled[(uint64_t)(c0 + k) * G + (n0 >> 3) + tid] = v;   // coalesced
        }
        __syncthreads();   // all waves done reading buf[b] before DMA overwrites it

        if (wave0 && (it + 2u < niter)) {
            tdm_load_tile(data + (uint64_t)(c0 + 2u * TM) * N + n0,
                          b * (uint32_t)TILE_BYTES, N);
        }
    }
}

// ---------------------------------------------------------------------------
// Kernel 2: padded gather. pooled (64MB) is L2-resident (192MB L2);
// output is write-once -> non-temporal stores keep pooled/idx in L2.
// ---------------------------------------------------------------------------
__global__ __launch_bounds__(256) void
OctreeGather_kernel(const float* __restrict__ pooled, const int* __restrict__ ci,
                    float* __restrict__ out, uint32_t G, uint32_t Mq, uint64_t total) {
    uint64_t i = (uint64_t)blockIdx.x * 256u + threadIdx.x;
    if (i >= total) return;
    uint32_t c  = (uint32_t)(i / Mq);
    uint32_t m4 = (uint32_t)(i - (uint64_t)c * Mq);

    i32x4 g4 = ((const i32x4*)ci)[m4];
    const float* prow = pooled + (uint64_t)c * G;
    f32x4 r;
    r.x = (g4.x >= 0) ? prow[g4.x] : 0.0f;
    r.y = (g4.y >= 0) ? prow[g4.y] : 0.0f;
    r.z = (g4.z >= 0) ? prow[g4.z] : 0.0f;
    r.w = (g4.w >= 0) ? prow[g4.w] : 0.0f;

    f32x4* dst = (f32x4*)(out + (uint64_t)c * (uint64_t)Mq * 4u) + m4;
    __builtin_nontemporal_store(r, dst);
}

// ---------------------------------------------------------------------------
// Fallback: fused direct gather-max (used only if d_ws < 64MB).
// ---------------------------------------------------------------------------
__device__ __forceinline__ float fused_max8(const float* row, int g) {
    if (g < 0) return 0.0f;
    const f32x4* p = (const f32x4*)(row + ((uint64_t)(uint32_t)g << 3));
    f32x4 a = __builtin_nontemporal_load(p);
    f32x4 b = __builtin_nontemporal_load(p + 1);
    return max8_v(a, b);
}

__global__ __launch_bounds__(256) void
OctreeFused_kernel(const float* __restrict__ data, const int* __restrict__ ci,
                   float* __restrict__ out, uint32_t N, uint32_t Mq, uint64_t total) {
    uint64_t i = (uint64_t)blockIdx.x * 256u + threadIdx.x;
    if (i >= total) return;
    uint32_t c  = (uint32_t)(i / Mq);
    uint32_t m4 = (uint32_t)(i - (uint64_t)c * Mq);

    i32x4 g4 = ((const i32x4*)ci)[m4];
    const float* row = data + (uint64_t)c * N;
    f32x4 r;
    r.x = fused_max8(row, g4.x);
    r.y = fused_max8(row, g4.y);
    r.z = fused_max8(row, g4.z);
    r.w = fused_max8(row, g4.w);

    f32x4* dst = (f32x4*)(out + (uint64_t)c * (uint64_t)Mq * 4u) + m4;
    __builtin_nontemporal_store(r, dst);
}

// ---------------------------------------------------------------------------
extern "C" void kernel_launch(void* const* d_in, const int* in_sizes, int n_in,
                              void* d_out, int out_size, void* d_ws, size_t ws_size,
                              hipStream_t stream) {
    const float* data = (const float*)d_in[0];
    const int*   ci   = (const int*)d_in[1];
    float*       out  = (float*)d_out;

    const uint32_t M = (uint32_t)in_sizes[1];                 // 262144
    const uint32_t C = (uint32_t)((uint64_t)out_size / M);    // 128
    const uint64_t N = (uint64_t)in_sizes[0] / C;             // 1048576
    const uint32_t G = (uint32_t)(N / 8u);                    // 131072
    const uint32_t Mq = M / 4u;

    const size_t pooled_bytes = (size_t)C * G * sizeof(float); // 64 MB
    const bool tiling_ok = (N % TK == 0) && (C % TM == 0) && (C / TM >= 2);

    const uint64_t total = (uint64_t)C * Mq;                  // gather threads
    const uint32_t gblocks = (uint32_t)((total + 255u) / 256u);

    if (ws_size >= pooled_bytes && tiling_ok) {
        float* pooled = (float*)d_ws;
        dim3 grid1((uint32_t)(N / TK));      // 512 workgroups
        OctreePool_tdm_kernel<<<grid1, dim3(256), 0, stream>>>(
            data, pooled, (uint32_t)N, G, C / TM);
        OctreeGather_kernel<<<dim3(gblocks), dim3(256), 0, stream>>>(
            pooled, ci, out, G, Mq, total);
    } else {
        OctreeFused_kernel<<<dim3(gblocks), dim3(256), 0, stream>>>(
            data, ci, out, (uint32_t)N, Mq, total);
    }
}